// AttentionEncoder_18279380811872
// MI455X (gfx1250) — compile-verified
//
#include <hip/hip_runtime.h>
#include <math.h>

// ---------------------------------------------------------------------------
// Fused self-attention (flash-style) for B=16, L=2048, D=256, fp32.
// out = softmax(fe @ fe^T) @ fe ; d_out = [out, 6 x out] (attentions).
// Matrix core: v_wmma_f32_16x16x4_f32 (exact fp32, matches fp32 reference).
// K/V tiles double-buffered in LDS via async global->LDS loads (ASYNCcnt).
// Epilogue: normalize into LDS, then coalesced b128 broadcast stores.
// ---------------------------------------------------------------------------

typedef float v2f __attribute__((ext_vector_type(2)));
typedef float v8f __attribute__((ext_vector_type(8)));

#define B_   16
#define L_   2048
#define D_   256
#define KT   16                 // keys per tile
#define NKT  (L_ / KT)          // 128 key-tile iterations
#define WAVES 4
#define ROWS_PER_WG (16 * WAVES)
#define OUT_STRIDE ((size_t)B_ * L_ * D_)   // 8,388,608 elements
#define NREP 7                  // out + 6 attention copies

#if __has_builtin(__builtin_amdgcn_global_load_async_to_lds_b128)
#define HAS_ASYNC 1
#else
#define HAS_ASYNC 0
#endif

// Builtin signature (from hipcc diagnostic): (AS1 v4i*, AS3 v4i*, imm, imm)
typedef int v4i_t __attribute__((vector_size(16)));
typedef __attribute__((address_space(1))) v4i_t* g128_t;
typedef __attribute__((address_space(3))) v4i_t* l128_t;

__device__ __forceinline__ void async_copy16(const float* g, float* l) {
#if HAS_ASYNC
  __builtin_amdgcn_global_load_async_to_lds_b128((g128_t)(void*)g,
                                                 (l128_t)(void*)l, 0, 0);
#else
  *(float4*)l = *(const float4*)g;
#endif
}

__device__ __forceinline__ void wait_async0() {
#if HAS_ASYNC
#if __has_builtin(__builtin_amdgcn_s_wait_asynccnt)
  __builtin_amdgcn_s_wait_asynccnt(0);
#else
  asm volatile("s_wait_asynccnt 0" ::: "memory");
#endif
#endif
}

__global__ __launch_bounds__(WAVES * 32)
void attn_fused_kernel(const float* __restrict__ fe, float* __restrict__ out) {
  // Manual union: 16384 floats = 64KB.
  //  During key loop : [0, 8192)  = double-buffered K tile (2 x 16KB)
  //                    [8192 + wv*256, +256) = per-wave P staging
  //  Epilogue        : [wv*4096, +4096) = per-wave O staging (16x256)
  __shared__ float smem[16384];

  const int tid  = threadIdx.x;
  const int wv   = tid >> 5;
  const int lane = tid & 31;
  const int l15  = lane & 15;
  const int hi   = lane >> 4;                  // 0: K0/K1 half, 1: K2/K3 half

  const int b     = blockIdx.x / (L_ / ROWS_PER_WG);
  const int tile  = blockIdx.x % (L_ / ROWS_PER_WG);
  const int qrow0 = tile * ROWS_PER_WG + wv * 16;

  const float* feb = fe + (size_t)b * L_ * D_;
  float* const ktile  = smem;                     // [2][KT*D_]
  float* const pldsw  = smem + 2 * KT * D_ + wv * 256;

  // ---- Q tile as 64 A-matrix fragments (16x4 f32 layout), kept in VGPRs ----
  v2f qf[64];
  {
    const float* qp = feb + (size_t)(qrow0 + l15) * D_ + 2 * hi;
#pragma unroll
    for (int i = 0; i < 64; ++i)
      qf[i] = *(const v2f*)(qp + 4 * i);
  }

  // ---- accumulators / online-softmax state ----
  v8f oacc[16];
#pragma unroll
  for (int c = 0; c < 16; ++c) oacc[c] = (v8f){};
  float rm[8], rl[8];
#pragma unroll
  for (int r = 0; r < 8; ++r) { rm[r] = -1e30f; rl[r] = 0.0f; }

  // ---- prologue: async-load key tile 0 into buffer 0 ----
  {
    const float* g = feb;                      // keys 0..15 (rows contiguous)
#pragma unroll
    for (int s = 0; s < 8; ++s)
      async_copy16(g + tid * 4 + s * 512, &ktile[tid * 4 + s * 512]);
    wait_async0();
    __syncthreads();
  }

  for (int j = 0; j < NKT; ++j) {
    const int cur = j & 1;

    // prefetch next key tile into the other buffer while we compute
    if (j + 1 < NKT) {
      const float* g = feb + (size_t)(j + 1) * KT * D_;
      float* lb = &ktile[(cur ^ 1) * (KT * D_)];
#pragma unroll
      for (int s = 0; s < 8; ++s)
        async_copy16(g + tid * 4 + s * 512, lb + tid * 4 + s * 512);
    }

    const float* kb = &ktile[cur * (KT * D_)];

    // ---- S = Q * K^T  (16x16), 64 x v_wmma_f32_16x16x4_f32 over D=256 ----
    v8f s = (v8f){};
    {
      const float* kp = kb + l15 * D_ + 2 * hi;  // B-frag of K^T == A-frag of K
#pragma unroll
      for (int i = 0; i < 64; ++i) {
        v2f bfr = *(const v2f*)(kp + 4 * i);
        s = __builtin_amdgcn_wmma_f32_16x16x4_f32(
            false, qf[i], false, bfr, (short)0, s, false, false);
      }
    }

    // ---- online softmax (row = r + 8*hi, col = l15 in C/D layout) ----
    float p[8];
#pragma unroll
    for (int r = 0; r < 8; ++r) {
      float x = s[r];
      float m = x;
      m = fmaxf(m, __shfl_xor(m, 1, 32));
      m = fmaxf(m, __shfl_xor(m, 2, 32));
      m = fmaxf(m, __shfl_xor(m, 4, 32));
      m = fmaxf(m, __shfl_xor(m, 8, 32));      // row max within 16-lane half
      float mnew  = fmaxf(rm[r], m);
      float alpha = __expf(rm[r] - mnew);
      float pe    = __expf(x - mnew);
      float su = pe;
      su += __shfl_xor(su, 1, 32);
      su += __shfl_xor(su, 2, 32);
      su += __shfl_xor(su, 4, 32);
      su += __shfl_xor(su, 8, 32);             // row sum
      rl[r] = rl[r] * alpha + su;
      rm[r] = mnew;
      p[r]  = pe;
#pragma unroll
      for (int c = 0; c < 16; ++c) oacc[c][r] *= alpha;
    }

    // ---- stage P through LDS, reload as A-layout fragments (K-dim = 16) ----
#pragma unroll
    for (int r = 0; r < 8; ++r)
      pldsw[(r + 8 * hi) * 16 + l15] = p[r];
    asm volatile("" ::: "memory");             // DS ops are in-order per wave
    v2f pa[4];
#pragma unroll
    for (int kk = 0; kk < 4; ++kk)
      pa[kk] = *(const v2f*)&pldsw[l15 * 16 + 4 * kk + 2 * hi];

    // ---- O += P * V  (V == K tile), 16 col-chunks x 4 WMMAs ----
#pragma unroll
    for (int c = 0; c < 16; ++c) {
#pragma unroll
      for (int kk = 0; kk < 4; ++kk) {
        int row = 4 * kk + 2 * hi;
        v2f bfr;
        bfr.x = kb[row * D_ + c * 16 + l15];
        bfr.y = kb[(row + 1) * D_ + c * 16 + l15];
        oacc[c] = __builtin_amdgcn_wmma_f32_16x16x4_f32(
            false, pa[kk], false, bfr, (short)0, oacc[c], false, false);
      }
    }

    // ---- make prefetched tile visible; retire reads before next overwrite --
    wait_async0();
    __syncthreads();
  }

  // ---- epilogue: normalize into LDS staging (16x256 per wave) ----
  float* const ostage = smem + wv * (16 * D_);
#pragma unroll
  for (int r = 0; r < 8; ++r) {
    float inv = 1.0f / rl[r];
    int rowb = (r + 8 * hi) * D_ + l15;
#pragma unroll
    for (int c = 0; c < 16; ++c)
      ostage[rowb + c * 16] = oacc[c][r] * inv;
  }
  asm volatile("" ::: "memory");               // DS in-order per wave

  // ---- coalesced b128 broadcast stores: out + 6 attention layers ----
  {
    const size_t gbase = ((size_t)b * L_ + qrow0) * D_;
    for (int i = 0; i < 32; ++i) {             // rolled: compact code
      int off = (i * 32 + lane) * 4;
      float4 v = *(const float4*)&ostage[off];
#pragma unroll
      for (int rep = 0; rep < NREP; ++rep)
        *(float4*)&out[gbase + off + (size_t)rep * OUT_STRIDE] = v;
    }
  }
}

extern "C" void kernel_launch(void* const* d_in, const int* in_sizes, int n_in,
                              void* d_out, int out_size, void* d_ws, size_t ws_size,
                              hipStream_t stream) {
  (void)in_sizes; (void)n_in; (void)d_ws; (void)ws_size; (void)out_size;
  const float* fe = (const float*)d_in[0];
  float* out = (float*)d_out;
  dim3 grid(B_ * (L_ / ROWS_PER_WG));   // 16 * 32 = 512 workgroups
  dim3 block(WAVES * 32);               // 4 waves (wave32)
  attn_fused_kernel<<<grid, block, 0, stream>>>(fe, out);
}